// NoisyTopKRouter_30167850287772
// MI455X (gfx1250) — compile-verified
//
#include <hip/hip_runtime.h>
#include <hip/hip_bf16.h>
#include <stdint.h>

typedef float v2f  __attribute__((ext_vector_type(2)));
typedef float v8f  __attribute__((ext_vector_type(8)));
typedef unsigned int u32x4 __attribute__((ext_vector_type(4)));
typedef int   i32x4 __attribute__((ext_vector_type(4)));
typedef int   i32x8 __attribute__((ext_vector_type(8)));

#define N_TOKENS 16384
#define DIM      2048
#define NE       64
#define EPS_F    1e-9f
#define KC       64
#define NCHUNK   (DIM / KC)

#if defined(__has_builtin)
#if __has_builtin(__builtin_amdgcn_tensor_load_to_lds) && \
    __has_builtin(__builtin_amdgcn_s_wait_tensorcnt)
#define USE_TDM 1
#endif
#endif
#ifndef USE_TDM
#define USE_TDM 0
#endif

__device__ __forceinline__ float softplus_f(float v) {
    return (v > 20.0f) ? v : log1pf(expf(v));
}

#if USE_TDM
// Issue one TDM load of a 16x64 f32 tile (row stride DIM in memory) into LDS
// at byte offset lds_off, with 16B of LDS padding after every 64-DWORD row
// (=> 68-float LDS row stride). Wave-level instruction; tracked on TENSORcnt.
__device__ __forceinline__ void tdm_load_tile(const float* gsrc, unsigned lds_off) {
    unsigned long long ga = (unsigned long long)(uintptr_t)gsrc;
    u32x4 g0;
    g0[0] = 1u;                                        // count=1, no gather
    g0[1] = lds_off;                                   // lds_addr (bytes)
    g0[2] = (unsigned)(ga & 0xffffffffull);            // global_addr[31:0]
    g0[3] = (unsigned)((ga >> 32) & 0x01ffffffull)     // global_addr[56:32]
          | (2u << 30);                                // type=2 ("image")
    i32x8 g1;
    g1[0] = (int)((2u << 16)     // data_size = 4 bytes
                | (1u << 20)     // pad_enable
                | (5u << 22)     // pad_interval: 64 DWORDs
                | (3u << 25));   // pad_amount:   4 DWORDs (16B)
    g1[1] = (int)((unsigned)DIM << 16);          // tensor_dim0[15:0] (=2048)
    g1[2] = (int)((unsigned)N_TOKENS << 16);     // dim0 hi=0 | tensor_dim1 lo
    g1[3] = (int)(64u << 16);                    // dim1 hi=0 | tile_dim0=64
    g1[4] = (int)16u;                            // tile_dim1=16, tile_dim2=0
    g1[5] = (int)DIM;                            // tensor_dim0_stride lo = 2048
    g1[6] = 0;                                   // stride0 hi | stride1 lo
    g1[7] = 0;                                   // stride1 hi
    i32x4 z4 = {0, 0, 0, 0};
#if __clang_major__ >= 23
    i32x8 z8 = {0, 0, 0, 0, 0, 0, 0, 0};
    __builtin_amdgcn_tensor_load_to_lds(g0, g1, z4, z4, z8, 0);
#else
    __builtin_amdgcn_tensor_load_to_lds(g0, g1, z4, z4, 0);
#endif
}
#endif

// Fused: dual GEMM (logits + noise-logits via f32 WMMA, TDM-staged x tiles)
// -> per-token noisy top-k routing -> deterministic per-WG partials.
__global__ __launch_bounds__(256) void moe_router_fused(
    const float* __restrict__ x, const float* __restrict__ noise,
    const float* __restrict__ Wg, const float* __restrict__ Wn,
    float* __restrict__ out, float* __restrict__ partials)
{
    __shared__ __align__(16) float xs[2][16 * 68];  // double-buffered x chunks
    __shared__ float lgt[16 * 68];                  // logits tile (16 tok x 64)
    __shared__ float nst[16 * 68];                  // noise-logit tile
    __shared__ float loadTok[16 * 64];
    __shared__ float impTok[16 * 64];

    const int tid  = threadIdx.x;
    const int lane = tid & 31;
    const int w    = tid >> 5;
    const int lo   = lane & 15;
    const int hi   = lane >> 4;
    const int mBase = blockIdx.x * 16;

    const float* Wsrc = (w < 4) ? Wg : Wn;
    const int ebase = (w & 3) * 16;

    v8f acc = {0.f, 0.f, 0.f, 0.f, 0.f, 0.f, 0.f, 0.f};

    const float* xTileBase = x + (size_t)mBase * DIM;

#if USE_TDM
    const unsigned xsOff[2] = { (unsigned)(uintptr_t)&xs[0][0],
                                (unsigned)(uintptr_t)&xs[1][0] };
    if (w == 0) {
        tdm_load_tile(xTileBase, xsOff[0]);
        __builtin_amdgcn_s_wait_tensorcnt((short)0);
    }
    __syncthreads();
#else
    const int xr = tid >> 4;
    const int xc = (tid & 15) * 4;
#endif

    for (int c = 0; c < NCHUNK; ++c) {
        const int k0 = c * KC;
#if USE_TDM
        // overlap: kick off chunk c+1 while computing chunk c
        if (w == 0 && c + 1 < NCHUNK) {
            tdm_load_tile(xTileBase + (size_t)(k0 + KC), xsOff[(c + 1) & 1]);
        }
#else
        float4 xv = *reinterpret_cast<const float4*>(
            xTileBase + (size_t)xr * DIM + k0 + xc);
        *reinterpret_cast<float4*>(&xs[c & 1][xr * 68 + xc]) = xv;
        if (k0 + KC < DIM) {
            __builtin_prefetch(xTileBase + (size_t)xr * DIM + k0 + KC + xc, 0, 1);
        }
        __syncthreads();
#endif
        // A frag: 16x4 f32 layout => lane(0..15)=row, VGPR0={K, K+2 hi lanes}
        const float* arow = &xs[c & 1][lo * 68 + 2 * hi];
        // B frag: B[k][n] = W[n][k]; lane lo = n, hi selects K pair (L2-hot)
        const float* wrow = Wsrc + (size_t)(ebase + lo) * DIM + k0 + 2 * hi;
        #pragma unroll
        for (int kk = 0; kk < KC; kk += 4) {
            v2f a = *reinterpret_cast<const v2f*>(arow + kk);
            v2f b = *reinterpret_cast<const v2f*>(wrow + kk);
            acc = __builtin_amdgcn_wmma_f32_16x16x4_f32(
                false, a, false, b, (short)0, acc, false, false);
        }
#if USE_TDM
        if (w == 0) __builtin_amdgcn_s_wait_tensorcnt((short)0);
#endif
        __syncthreads();
    }

    // Scatter 16x16 f32 C tile to LDS: VGPR r -> row r + 8*hi, col = lo
    float* ctile = (w < 4) ? lgt : nst;
    #pragma unroll
    for (int r = 0; r < 8; ++r) {
        ctile[(r + 8 * hi) * 68 + ebase + lo] = acc[r];
    }
    __syncthreads();

    // ---------------- routing epilogue: wave w handles tokens 2w, 2w+1 ------
    for (int s = 0; s < 2; ++s) {
        const int tloc = 2 * w + s;
        const int t = mBase + tloc;
        const int ea = lane, eb = lane + 32;

        float lgA = lgt[tloc * 68 + ea];
        float lgB = lgt[tloc * 68 + eb];
        float stA = softplus_f(nst[tloc * 68 + ea]) + EPS_F;
        float stB = softplus_f(nst[tloc * 68 + eb]) + EPS_F;
        float nzA = lgA + noise[(size_t)t * NE + ea] * stA;
        float nzB = lgB + noise[(size_t)t * NE + eb] * stB;

        // iterative top-3 (argmax + remove), tie-break: lower index first
        float wa = nzA, wb = nzB;
        float tv[3]; int ti[3];
        #pragma unroll
        for (int j = 0; j < 3; ++j) {
            float lv; int li;
            if (wb > wa) { lv = wb; li = eb; } else { lv = wa; li = ea; }
            #pragma unroll
            for (int off = 16; off > 0; off >>= 1) {
                float ov = __shfl_xor(lv, off, 32);
                int   oi = __shfl_xor(li, off, 32);
                if (ov > lv || (ov == lv && oi < li)) { lv = ov; li = oi; }
            }
            tv[j] = lv; ti[j] = li;
            if (ea == li) wa = -INFINITY;
            if (eb == li) wb = -INFINITY;
        }

        float d  = expf(tv[1] - tv[0]);
        float g0 = 1.0f / (1.0f + d);
        float g1 = d / (1.0f + d);

        float kthA = (ea == ti[0] || ea == ti[1]) ? tv[2] : tv[1];
        float kthB = (eb == ti[0] || eb == ti[1]) ? tv[2] : tv[1];
        float zA = (lgA - kthA) / (stA + EPS_F);
        float zB = (lgB - kthB) / (stB + EPS_F);
        const float inv_sqrt2 = 0.70710678118654752f;
        float ldA = 0.5f * (1.0f + erff(zA * inv_sqrt2));
        float ldB = 0.5f * (1.0f + erff(zB * inv_sqrt2));

        loadTok[tloc * 64 + ea] = ldA;
        loadTok[tloc * 64 + eb] = ldB;
        impTok[tloc * 64 + ea] = (ea == ti[0]) ? g0 : ((ea == ti[1]) ? g1 : 0.0f);
        impTok[tloc * 64 + eb] = (eb == ti[0]) ? g0 : ((eb == ti[1]) ? g1 : 0.0f);

        if (lane == 0) {
            out[2 * t + 0] = (float)ti[0];
            out[2 * t + 1] = (float)ti[1];
            out[2 * N_TOKENS + 2 * t + 0] = g0;
            out[2 * N_TOKENS + 2 * t + 1] = g1;
            out[4 * N_TOKENS + t] = g0;   // priority = max weight (g0 >= g1)
        }
    }
    __syncthreads();

    if (tid < 64) {
        float impS = 0.f, ldS = 0.f;
        #pragma unroll
        for (int tt = 0; tt < 16; ++tt) {
            impS += impTok[tt * 64 + tid];
            ldS  += loadTok[tt * 64 + tid];
        }
        partials[(size_t)blockIdx.x * 128 + tid]      = impS;
        partials[(size_t)blockIdx.x * 128 + 64 + tid] = ldS;
    }
}

__global__ __launch_bounds__(64) void moe_router_finalize(
    const float* __restrict__ partials, float* __restrict__ out)
{
    __shared__ float sImp[64];
    __shared__ float sLoad[64];
    const int e = threadIdx.x;
    float impS = 0.f, ldS = 0.f;
    for (int g = 0; g < N_TOKENS / 16; ++g) {
        impS += partials[(size_t)g * 128 + e];
        ldS  += partials[(size_t)g * 128 + 64 + e];
    }
    sImp[e] = impS;
    sLoad[e] = ldS;
    __syncthreads();
    if (e == 0) {
        float mi = 0.f, ml = 0.f;
        for (int i = 0; i < 64; ++i) { mi += sImp[i]; ml += sLoad[i]; }
        mi *= (1.0f / 64.0f); ml *= (1.0f / 64.0f);
        float vi = 0.f, vl = 0.f;
        for (int i = 0; i < 64; ++i) {
            float a = sImp[i] - mi, b = sLoad[i] - ml;
            vi += a * a; vl += b * b;
        }
        vi *= (1.0f / 64.0f); vl *= (1.0f / 64.0f);
        float aux = 0.1f * (vi / (mi * mi + EPS_F))
                  + 0.1f * (vl / (ml * ml + EPS_F));
        out[5 * N_TOKENS] = aux;
    }
}

extern "C" void kernel_launch(void* const* d_in, const int* in_sizes, int n_in,
                              void* d_out, int out_size, void* d_ws, size_t ws_size,
                              hipStream_t stream) {
    const float* x       = (const float*)d_in[0];
    const float* noise   = (const float*)d_in[1];
    const float* W_gate  = (const float*)d_in[2];
    const float* W_noise = (const float*)d_in[3];
    float* out = (float*)d_out;
    float* partials = (float*)d_ws;   // 1024 * 128 floats = 512 KB

    (void)in_sizes; (void)n_in; (void)out_size; (void)ws_size;

    moe_router_fused<<<N_TOKENS / 16, 256, 0, stream>>>(
        x, noise, W_gate, W_noise, out, partials);
    moe_router_finalize<<<1, 64, 0, stream>>>(partials, out);
}